// Loss_69887707841277
// MI455X (gfx1250) — compile-verified
//
#include <hip/hip_runtime.h>
#include <math.h>

#define BB      2
#define NN      4096
#define NPOINT  204          // int(4096*0.05)
#define NP      5
#define NG      (BB*NPOINT)  // 408 groups
#define NPAD_MAX 416
#define EPSV    0.1f
#define FPS_T   1024
#define FPS_K   (NN / FPS_T) // 4

typedef float v2f __attribute__((ext_vector_type(2)));
typedef float v8f __attribute__((ext_vector_type(8)));

struct Params {
    int   ns[NP];      // nsample per percentage
    int   npad[NP];    // nsample padded to multiple of 16
    int   off[NP];     // int offset into idx buffer
    float r2[NP];      // squared radius = p
    float wfin[NP];    // (p*100)^2 / (G*5)
    float el;          // expect_len = sqrt(pi/N)
};

__global__ void zero_kernel(float* out) { out[0] = 0.0f; }

// ---------------- Furthest point sampling: one block per batch ----------------
__global__ __launch_bounds__(FPS_T)
void fps_kernel(const float* __restrict__ pcd, float* __restrict__ new_xyz) {
    const int b = blockIdx.x;
    const int t = threadIdx.x;
    const float* pts = pcd + (size_t)b * NN * 3;

    float px[FPS_K], py[FPS_K], pz[FPS_K], dist[FPS_K];
#pragma unroll
    for (int k = 0; k < FPS_K; ++k) {
        int i = t + k * FPS_T;
        px[k] = pts[i*3+0]; py[k] = pts[i*3+1]; pz[k] = pts[i*3+2];
        dist[k] = 1e10f;
    }

    __shared__ float s_val[FPS_T];
    __shared__ int   s_idx[FPS_T];
    __shared__ float s_last[3];

    if (t == 0) {
        s_last[0] = pts[0]; s_last[1] = pts[1]; s_last[2] = pts[2];
        new_xyz[(b*NPOINT+0)*3+0] = pts[0];
        new_xyz[(b*NPOINT+0)*3+1] = pts[1];
        new_xyz[(b*NPOINT+0)*3+2] = pts[2];
    }
    __syncthreads();

    for (int s = 1; s < NPOINT; ++s) {
        float lx = s_last[0], ly = s_last[1], lz = s_last[2];
        float bv = -1.0f; int bi = 0;
#pragma unroll
        for (int k = 0; k < FPS_K; ++k) {
            float dx = px[k]-lx, dy = py[k]-ly, dz = pz[k]-lz;
            float d = dx*dx + dy*dy + dz*dz;
            dist[k] = fminf(dist[k], d);
            int i = t + k * FPS_T;
            if (dist[k] > bv || (dist[k] == bv && i < bi)) { bv = dist[k]; bi = i; }
        }
        s_val[t] = bv; s_idx[t] = bi;
        __syncthreads();
        for (int stride = FPS_T/2; stride > 0; stride >>= 1) {
            if (t < stride) {
                float v2 = s_val[t+stride]; int i2 = s_idx[t+stride];
                if (v2 > s_val[t] || (v2 == s_val[t] && i2 < s_idx[t])) {
                    s_val[t] = v2; s_idx[t] = i2;
                }
            }
            __syncthreads();
        }
        if (t == 0) {
            int nxt = s_idx[0];
            float x = pts[nxt*3+0], y = pts[nxt*3+1], z = pts[nxt*3+2];
            s_last[0]=x; s_last[1]=y; s_last[2]=z;
            new_xyz[(b*NPOINT+s)*3+0]=x;
            new_xyz[(b*NPOINT+s)*3+1]=y;
            new_xyz[(b*NPOINT+s)*3+2]=z;
        }
        __syncthreads();
    }
}

// ---------------- Ball query: one wave32 per (query, batch, percentage) ------
__global__ __launch_bounds__(32)
void ballq_kernel(const float* __restrict__ pcd, const float* __restrict__ new_xyz,
                  int* __restrict__ idx, Params P) {
    const int m = blockIdx.x, b = blockIdx.y, p = blockIdx.z;
    const int lane = threadIdx.x;
    const int ns = P.ns[p];
    const float r2 = P.r2[p];
    const int base = P.off[p] + (b*NPOINT + m) * ns;

    const float* q = new_xyz + (b*NPOINT + m)*3;
    const float qx = q[0], qy = q[1], qz = q[2];
    const float* pts = pcd + (size_t)b * NN * 3;

    int found = 0, first = -1;
    for (int c = 0; c < NN && found < ns; c += 32) {
        int i = c + lane;
        float dx = pts[i*3+0]-qx, dy = pts[i*3+1]-qy, dz = pts[i*3+2]-qz;
        bool in = (dx*dx + dy*dy + dz*dz) < r2;
        unsigned mask = (unsigned)__ballot(in);
        if (first < 0 && mask) first = c + __ffs(mask) - 1;
        int pos = found + __popc(mask & ((1u << lane) - 1u));
        if (in && pos < ns) idx[base + pos] = i;
        found += __popc(mask);
    }
    if (first < 0) first = 0;
    int total = found < ns ? found : ns;
    for (int k = total + lane; k < ns; k += 32) idx[base + k] = first; // pn2 pad
}

// ------- Per-group NN distance via WMMA Gram matrix + loss accumulation ------
// d2 = |g_i|^2 + |g_j|^2 - 2*G.G^T  (same formulation as the reference).
// sqrt/max are monotone, so we reduce raw d2 and apply sqrt(max(.,0)) once
// per row AFTER the reduction. The diagonal tile (tj == ti) is peeled and
// masked with a per-lane constant additive mask; off-diagonal tiles are
// covered by two contiguous select-free loops ([ti+1,ntiles) and [0,ti)).
__global__ __launch_bounds__(256)
void nn_kernel(const float* __restrict__ pcd, const int* __restrict__ idx,
               float* __restrict__ out, Params P) {
    const int g = blockIdx.x;            // group in [0, NG)
    const int p = blockIdx.y;            // percentage
    const int b = g / NPOINT;
    const int ns = P.ns[p], npad = P.npad[p];
    const int base = P.off[p] + g * ns;
    const int t = threadIdx.x;

    __shared__ float4 s_pts[NPAD_MAX];   // x, y, z, |x|^2
    __shared__ float  s_acc;
    if (t == 0) s_acc = 0.0f;

    for (int row = t; row < npad; row += 256) {
        float x, y, z;
        if (row < ns) {
            int i = idx[base + row];
            const float* pp = pcd + ((size_t)b * NN + i) * 3;
            x = pp[0]; y = pp[1]; z = pp[2];
        } else {                          // far sentinel: never a nearest neighbor
            x = 1.0e4f + (float)row; y = 0.0f; z = 0.0f;
        }
        s_pts[row] = make_float4(x, y, z, x*x + y*y + z*z);
    }
    __syncthreads();

    const int wave = t >> 5, lane = t & 31;
    const int half = lane >> 4, l16 = lane & 15;
    const int ntiles = npad >> 4;
    const int dloc = l16 - 8*half;       // v-index of this lane's diagonal element

    // Per-lane constant additive mask for the diagonal tile (computed once).
    float dmask[8];
#pragma unroll
    for (int v = 0; v < 8; ++v) dmask[v] = (v == dloc) ? 3.4e38f : 0.0f;

    for (int ti = wave; ti < ntiles; ti += 8) {
        // A tile (16x4): lanes 0-15 -> K=0,1 (x,y); lanes 16-31 -> K=2,3 (z,0)
        v2f a;
        {
            float4 pr = s_pts[ti*16 + l16];
            a.x = half ? pr.z : pr.x;
            a.y = half ? 0.0f : pr.y;
        }
        float sqa[8];
#pragma unroll
        for (int v = 0; v < 8; ++v) sqa[v] = s_pts[ti*16 + v + 8*half].w;

        float rmin[8];                    // running min of raw d2 per row
#pragma unroll
        for (int v = 0; v < 8; ++v) rmin[v] = 3.4e38f;

        // Branch/select-free off-diagonal tile body.
        auto body = [&](int tj) {
            float4 pc = s_pts[tj*16 + l16];
            v2f bmat;
            bmat.x = half ? pc.z : pc.x;
            bmat.y = half ? 0.0f : pc.y;
            float sqb = pc.w;                         // |col|^2 for col = l16

            v8f c = {0.f,0.f,0.f,0.f,0.f,0.f,0.f,0.f};
            c = __builtin_amdgcn_wmma_f32_16x16x4_f32(
                    false, a, false, bmat, (short)0, c, false, false);
#pragma unroll
            for (int v = 0; v < 8; ++v) {
                float d2 = fmaf(-2.0f, c[v], sqa[v] + sqb);
                rmin[v] = fminf(rmin[v], d2);
            }
        };

        // ---- peeled diagonal tile (tj == ti): additive self-distance mask ----
        {
            v8f c = {0.f,0.f,0.f,0.f,0.f,0.f,0.f,0.f};
            c = __builtin_amdgcn_wmma_f32_16x16x4_f32(
                    false, a, false, a, (short)0, c, false, false);
            float sqb = s_pts[ti*16 + l16].w;
#pragma unroll
            for (int v = 0; v < 8; ++v) {
                float d2 = fmaf(-2.0f, c[v], sqa[v] + sqb);
                rmin[v] = fminf(rmin[v], d2 + dmask[v]);   // diag -> ~inf
            }
        }

        // ---- off-diagonal tiles: two contiguous ranges, no wrap select ----
        for (int tj = ti + 1; tj < ntiles; ++tj) body(tj);
        for (int tj = 0;      tj < ti;     ++tj) body(tj);

        // min across the 16-lane half (rows live per-half); xor 1..8 stays in half
#pragma unroll
        for (int off = 1; off < 16; off <<= 1)
#pragma unroll
            for (int v = 0; v < 8; ++v)
                rmin[v] = fminf(rmin[v], __shfl_xor(rmin[v], off, 32));

        if (l16 == 0) {                  // lanes 0 and 16 own rows [ti*16+8*half .. +7]
            float local = 0.0f;
#pragma unroll
            for (int v = 0; v < 8; ++v) {
                int grow = ti*16 + v + 8*half;
                if (grow < ns) {
                    float nn = sqrtf(fmaxf(rmin[v], 0.0f));  // sqrt hoisted here
                    local += fabsf(nn + EPSV);
                }
            }
            atomicAdd(&s_acc, local);
        }
    }
    __syncthreads();

    if (t == 0) {
        float u = s_acc / (float)ns;                 // mean |nn + e| over group
        float d = u - P.el;
        atomicAdd(out, (d * d / (P.el + EPSV)) * P.wfin[p]);
    }
}

extern "C" void kernel_launch(void* const* d_in, const int* in_sizes, int n_in,
                              void* d_out, int out_size, void* d_ws, size_t ws_size,
                              hipStream_t stream) {
    const float* pcd = (const float*)d_in[0];
    float* out = (float*)d_out;

    float* ws      = (float*)d_ws;
    float* new_xyz = ws;                    // BB*NPOINT*3 = 1224 floats
    int*   idx     = (int*)(ws + 2048);     // aligned past new_xyz

    static const double PCT[NP] = {0.02, 0.04, 0.06, 0.08, 0.10};
    Params P;
    int off = 0;
    for (int p = 0; p < NP; ++p) {
        int ns    = (int)((double)NN * PCT[p]);      // 81,163,245,327,409
        P.ns[p]   = ns;
        P.npad[p] = (ns + 15) & ~15;                 // 96,176,256,336,416
        P.off[p]  = off;
        off      += NG * ns;
        P.r2[p]   = (float)PCT[p];                   // r^2 = p * RADIUS^2
        double w  = (PCT[p] * 100.0) * (PCT[p] * 100.0);
        P.wfin[p] = (float)(w / ((double)NG * (double)NP));
    }
    P.el = (float)sqrt(M_PI / (double)NN);           // expect_len (p-independent)

    zero_kernel<<<1, 1, 0, stream>>>(out);
    fps_kernel<<<dim3(BB), FPS_T, 0, stream>>>(pcd, new_xyz);
    ballq_kernel<<<dim3(NPOINT, BB, NP), 32, 0, stream>>>(pcd, new_xyz, idx, P);
    nn_kernel<<<dim3(NG, NP), 256, 0, stream>>>(pcd, idx, out, P);
}